// Net_24790551233195
// MI455X (gfx1250) — compile-verified
//
#include <hip/hip_runtime.h>
#include <hip/hip_bf16.h>
#include <cmath>

#define ALPHA  0.1f
#define THETA  0.5f
#define IN_F   50
#define HID    256
#define OUT_F  121
#define NLAYER 4

typedef __attribute__((ext_vector_type(2))) float v2f;
typedef __attribute__((ext_vector_type(8))) float v8f;

#define WMMA_F32(a, b, c) \
    __builtin_amdgcn_wmma_f32_16x16x4_f32(false, (a), false, (b), (short)0, (c), false, false)

// ---------------------------------------------------------------------------
// lin0: x0 = relu(x @ lin0_w + lin0_b); also h = x0.
// M=nrows, K=50, N=256. Block = 16 rows, 8 waves, 2 N-tiles/wave.
// K loop peeled: 12 unguarded steps (K 0..47) + 1 masked step (K 48..51).
// ---------------------------------------------------------------------------
#define L0_LDST 68   // 68 % 64 == 4 -> rows land on distinct banks
__global__ __launch_bounds__(256) void gemm_lin0(
    const float* __restrict__ x, const float* __restrict__ w,
    const float* __restrict__ b, float* __restrict__ x0,
    float* __restrict__ h, int nrows)
{
    __shared__ float As[16 * L0_LDST];
    const int tid = threadIdx.x;
    const int m0  = blockIdx.x * 16;

    // zero pad columns 50..53 (peeled step reads up to col 51)
    for (int idx = tid; idx < 16 * 4; idx += 256) {
        int r = idx >> 2, c = 50 + (idx & 3);
        As[r * L0_LDST + c] = 0.0f;
    }
    for (int idx = tid; idx < 16 * IN_F; idx += 256) {
        int r = idx / IN_F, c = idx % IN_F;
        int gm = m0 + r;
        As[r * L0_LDST + c] = (gm < nrows) ? x[(size_t)gm * IN_F + c] : 0.0f;
    }
    __syncthreads();

    const int wave  = tid >> 5;        // wave32 on gfx1250
    const int lane  = tid & 31;
    const int mrow  = lane & 15;
    const int khalf = (lane >> 4) * 2;
    const float kmask = (khalf == 0) ? 1.0f : 0.0f;  // rows 50/51 are zero

    for (int nt = wave; nt < 16; nt += 8) {
        const int n0 = nt * 16;
        v8f acc = {};
        for (int k = 0; k < 48; k += 4) {          // kr max = 44+2+1 = 47 < 50
            v2f a, bb;
            a.x = As[mrow * L0_LDST + k + khalf];
            a.y = As[mrow * L0_LDST + k + khalf + 1];
            const float* Bp = w + (size_t)(k + khalf) * HID + n0 + mrow;
            bb.x = Bp[0];
            bb.y = Bp[HID];
            acc = WMMA_F32(a, bb, acc);
        }
        {   // peeled step: K rows 48,49 valid (khalf==0), rows 50,51 zero
            v2f a, bb;
            a.x = As[mrow * L0_LDST + 48 + khalf];
            a.y = As[mrow * L0_LDST + 48 + khalf + 1];
            const float* Bp = w + (size_t)48 * HID + n0 + mrow;  // rows 48,49
            bb.x = kmask * Bp[0];
            bb.y = kmask * Bp[HID];
            acc = WMMA_F32(a, bb, acc);
        }
        const int crow = (lane >> 4) * 8;
        const int ccol = n0 + (lane & 15);
        const float bias = b[ccol];
        for (int j = 0; j < 8; ++j) {
            int gm = m0 + crow + j;
            if (gm < nrows) {
                float v = acc[j] + bias;
                v = v > 0.0f ? v : 0.0f;
                size_t idx = (size_t)gm * HID + ccol;
                x0[idx] = v;
                h[idx]  = v;
            }
        }
    }
}

// ---------------------------------------------------------------------------
// agg = ALPHA * x0   (folds the initial-residual term; scatter adds (1-a)*w*h)
// ---------------------------------------------------------------------------
__global__ __launch_bounds__(256) void init_agg(
    const float* __restrict__ x0, float* __restrict__ agg, int total)
{
    int i = blockIdx.x * 256 + threadIdx.x;
    if (i < total) agg[i] = ALPHA * x0[i];
}

// ---------------------------------------------------------------------------
// SpMM scatter: agg[dst] += (1-ALPHA) * w_e * h[src]. One wave per edge,
// 8 features per lane, hardware f32 atomics (GLOBAL_ATOMIC_ADD_F32).
// ---------------------------------------------------------------------------
__global__ __launch_bounds__(256) void scatter_edges(
    const int* __restrict__ ei, const float* __restrict__ ew,
    const float* __restrict__ h, float* __restrict__ agg,
    int nedges, int nnodes)
{
    int e = blockIdx.x * 8 + (threadIdx.x >> 5);
    if (e >= nedges) return;
    int lane = threadIdx.x & 31;
    int src = ei[e];
    int dst = ei[nedges + e];
    if ((unsigned)src >= (unsigned)nnodes || (unsigned)dst >= (unsigned)nnodes) return;
    float w = (1.0f - ALPHA) * ew[e];
    const float* hs = h + (size_t)src * HID;
    float* ad = agg + (size_t)dst * HID;
    #pragma unroll
    for (int j = 0; j < 8; ++j) {
        int f = lane + j * 32;
        unsafeAtomicAdd(&ad[f], w * hs[f]);
    }
}

// ---------------------------------------------------------------------------
// hidden layer GEMM + fused epilogue:
//   t = s @ W ;  h = relu((1-beta)*s + beta*t + h)   (in-place h update)
// Block = 64 rows x 256 cols. 8 waves; each wave: 2 N-tiles x 4 M-tiles
// (8 f32 accumulators, B loaded once per K-step, reused across 4 M-tiles).
// A staged in LDS (64 x 260 f32 = 66.5 KB), stride 260 breaks bank conflicts.
// ---------------------------------------------------------------------------
#define H_LDST 260
#define MB     64
__global__ __launch_bounds__(256) void gemm_hidden(
    const float* __restrict__ s, const float* __restrict__ W,
    float* __restrict__ h, int nrows, float beta)
{
    __shared__ float As[MB * H_LDST];
    const int tid = threadIdx.x;
    const int m0  = blockIdx.x * MB;

    for (int i = 0; i < MB; ++i) {
        int gm = m0 + i;
        As[i * H_LDST + tid] = (gm < nrows) ? s[(size_t)gm * HID + tid] : 0.0f;
    }
    __syncthreads();

    const int wave  = tid >> 5;
    const int lane  = tid & 31;
    const int mrow  = lane & 15;
    const int khalf = (lane >> 4) * 2;
    const int n0    = wave * 32;            // this wave's 2 N-tiles: n0, n0+16

    v8f acc[4][2];
    #pragma unroll
    for (int mi = 0; mi < 4; ++mi) {
        acc[mi][0] = (v8f){};
        acc[mi][1] = (v8f){};
    }

    const float* B0 = W + n0 + mrow;
    for (int k = 0; k < HID; k += 4) {
        const float* Bp = B0 + (size_t)(k + khalf) * HID;
        v2f b0, b1;
        b0.x = Bp[0];        b0.y = Bp[HID];
        b1.x = Bp[16];       b1.y = Bp[HID + 16];
        #pragma unroll
        for (int mi = 0; mi < 4; ++mi) {
            v2f a;
            a.x = As[(mi * 16 + mrow) * H_LDST + k + khalf];
            a.y = As[(mi * 16 + mrow) * H_LDST + k + khalf + 1];
            acc[mi][0] = WMMA_F32(a, b0, acc[mi][0]);
            acc[mi][1] = WMMA_F32(a, b1, acc[mi][1]);
        }
    }

    const int crow = (lane >> 4) * 8;
    #pragma unroll
    for (int mi = 0; mi < 4; ++mi) {
        #pragma unroll
        for (int ni = 0; ni < 2; ++ni) {
            const int ccol = n0 + ni * 16 + (lane & 15);
            #pragma unroll
            for (int j = 0; j < 8; ++j) {
                int lr = mi * 16 + crow + j;
                int gm = m0 + lr;
                if (gm < nrows) {
                    float sval = As[lr * H_LDST + ccol];
                    float outv = (1.0f - beta) * sval + beta * acc[mi][ni][j];
                    size_t idx = (size_t)gm * HID + ccol;
                    float r = outv + h[idx];
                    h[idx] = r > 0.0f ? r : 0.0f;
                }
            }
        }
    }
}

// ---------------------------------------------------------------------------
// lin1: out = h @ lin1_w + lin1_b.  M=nrows, K=256, N=121.
// Block = 64 rows. 8 waves, each: 1 N-tile x 4 M-tiles. Out-of-range columns
// are CLAMPED (not branched): a lane's B column only feeds the D column that
// same lane stores, and cols >= 121 are never stored.
// ---------------------------------------------------------------------------
__global__ __launch_bounds__(256) void gemm_lin1(
    const float* __restrict__ h, const float* __restrict__ w,
    const float* __restrict__ b, float* __restrict__ out, int nrows)
{
    __shared__ float As[MB * H_LDST];
    const int tid = threadIdx.x;
    const int m0  = blockIdx.x * MB;

    for (int i = 0; i < MB; ++i) {
        int gm = m0 + i;
        As[i * H_LDST + tid] = (gm < nrows) ? h[(size_t)gm * HID + tid] : 0.0f;
    }
    __syncthreads();

    const int wave  = tid >> 5;
    const int lane  = tid & 31;
    const int mrow  = lane & 15;
    const int khalf = (lane >> 4) * 2;
    const int n0    = wave * 16;            // 8 waves cover cols 0..127
    const int nn    = n0 + mrow;
    const int nc    = (nn < OUT_F) ? nn : (OUT_F - 1);   // clamped column

    v8f acc[4];
    #pragma unroll
    for (int mi = 0; mi < 4; ++mi) acc[mi] = (v8f){};

    for (int k = 0; k < HID; k += 4) {
        const int kr = k + khalf;
        v2f bb;
        bb.x = w[(size_t)kr * OUT_F + nc];
        bb.y = w[(size_t)(kr + 1) * OUT_F + nc];
        #pragma unroll
        for (int mi = 0; mi < 4; ++mi) {
            v2f a;
            a.x = As[(mi * 16 + mrow) * H_LDST + k + khalf];
            a.y = As[(mi * 16 + mrow) * H_LDST + k + khalf + 1];
            acc[mi] = WMMA_F32(a, bb, acc[mi]);
        }
    }

    const int crow = (lane >> 4) * 8;
    const int ccol = n0 + (lane & 15);
    const float bias = (ccol < OUT_F) ? b[ccol] : 0.0f;
    #pragma unroll
    for (int mi = 0; mi < 4; ++mi) {
        #pragma unroll
        for (int j = 0; j < 8; ++j) {
            int gm = m0 + mi * 16 + crow + j;
            if (gm < nrows && ccol < OUT_F)
                out[(size_t)gm * OUT_F + ccol] = acc[mi][j] + bias;
        }
    }
}

// ---------------------------------------------------------------------------
extern "C" void kernel_launch(void* const* d_in, const int* in_sizes, int n_in,
                              void* d_out, int out_size, void* d_ws, size_t ws_size,
                              hipStream_t stream) {
    const float* x    = (const float*)d_in[0];
    const int*   ei   = (const int*)  d_in[1];
    const float* ew   = (const float*)d_in[2];
    const float* l0w  = (const float*)d_in[3];
    const float* l0b  = (const float*)d_in[4];
    const float* Wall = (const float*)d_in[5];
    const float* l1w  = (const float*)d_in[6];
    const float* l1b  = (const float*)d_in[7];
    float* out = (float*)d_out;

    const int n  = in_sizes[0] / IN_F;   // 50000
    const int ne = in_sizes[2];          // 320000

    float* x0  = (float*)d_ws;
    float* h   = x0 + (size_t)n * HID;
    float* agg = h  + (size_t)n * HID;

    const int total = n * HID;

    gemm_lin0<<<(n + 15) / 16, 256, 0, stream>>>(x, l0w, l0b, x0, h, n);

    for (int l = 0; l < NLAYER; ++l) {
        float beta = logf(THETA / (float)(l + 1) + 1.0f);
        init_agg<<<(total + 255) / 256, 256, 0, stream>>>(x0, agg, total);
        scatter_edges<<<(ne + 7) / 8, 256, 0, stream>>>(ei, ew, h, agg, ne, n);
        gemm_hidden<<<(n + MB - 1) / MB, 256, 0, stream>>>(
            agg, Wall + (size_t)l * HID * HID, h, n, beta);
    }

    gemm_lin1<<<(n + MB - 1) / MB, 256, 0, stream>>>(h, l1w, l1b, out, n);
}